// MLA_56667798503608
// MI455X (gfx1250) — compile-verified
//
#include <hip/hip_runtime.h>

// ---------------- problem constants ----------------
#define B_      32
#define S_      4
#define T_      4096
#define START_  4092
#define DIM_    2048
#define H_      16
#define KVR_    512
#define DN_     128
#define DR_     64
#define DV_     128
#define CAT_    576            // KVR + DR (concat kv||pe)
#define QROW_   192            // DN + DR per head
#define NEG_    (-1000000000.0f)

// flash-decode tiling
#define NCHUNK_  8
#define CHUNK_T_ 512
#define TILE_    16
#define KPITCH_  580           // 576 + 4: rows 16B-aligned for B128 async, pitch%64==4 -> conflict-free QK reads
#define F4_PER_TILE_ (TILE_ * CAT_ / 4)   // 2304 float4 per tile
#define OPS_PER_WAVE_ 18                  // 2304 / 128 threads async ops per wave per tile

// ---------------- workspace layout (floats) ----------------
constexpr size_t QT_OFF    = 0;                                       // [128][576][16]
constexpr size_t NKV_OFF   = QT_OFF    + (size_t)B_*S_*CAT_*H_;       // [128][512]
constexpr size_t NPE_OFF   = NKV_OFF   + (size_t)B_*S_*KVR_;          // [128][64]
constexpr size_t OPART_OFF = NPE_OFF   + (size_t)B_*S_*DR_;           // [128][8][16][512]
constexpr size_t ML_OFF    = OPART_OFF + (size_t)B_*S_*NCHUNK_*H_*KVR_;// [128][8][16][2]
constexpr size_t OPROJ_OFF = ML_OFF    + (size_t)B_*S_*NCHUNK_*H_*2;  // [128][2048]

typedef float v2f __attribute__((ext_vector_type(2)));
typedef float v8f __attribute__((ext_vector_type(8)));

__device__ __forceinline__ v8f wmma4(v2f a, v2f b, v8f c) {
  // D = A(16x4) * B(4x16) + C(16x16), fp32
  return __builtin_amdgcn_wmma_f32_16x16x4_f32(false, a, false, b,
                                               (short)0, c, false, false);
}

// -------- CDNA5 async global->LDS path (ASYNCcnt), with safe fallback --------
#if __has_builtin(__builtin_amdgcn_global_load_async_to_lds_b128) && \
    __has_builtin(__builtin_amdgcn_s_wait_asynccnt)
#define HAVE_ASYNC_LDS 1
#else
#define HAVE_ASYNC_LDS 0
#endif

#if HAVE_ASYNC_LDS
typedef int v4i_ __attribute__((vector_size(4 * sizeof(int))));
typedef __attribute__((address_space(1))) v4i_* gptr4_;
typedef __attribute__((address_space(3))) v4i_* lptr4_;
#endif

__device__ __forceinline__ void copy16_lds(const float* __restrict__ g, float* l) {
#if HAVE_ASYNC_LDS
  __builtin_amdgcn_global_load_async_to_lds_b128((gptr4_)g, (lptr4_)l, 0, 0);
#else
  *reinterpret_cast<float4*>(l) = *reinterpret_cast<const float4*>(g);
#endif
}

__device__ __forceinline__ void wait_async(bool pending) {
#if HAVE_ASYNC_LDS
  if (pending) __builtin_amdgcn_s_wait_asynccnt(OPS_PER_WAVE_);
  else         __builtin_amdgcn_s_wait_asynccnt(0);
#else
  (void)pending;
#endif
}

// ============ Kernel A: kv_a projection + rmsnorm + rope(k_pe) ============
__global__ void mla_kv_prep(const float* __restrict__ x,
                            const float* __restrict__ wkv_a,
                            const float* __restrict__ kvnw,
                            const float* __restrict__ fcos,
                            const float* __restrict__ fsin,
                            float* __restrict__ nkv,
                            float* __restrict__ npe) {
  __shared__ float xs[DIM_];
  __shared__ float kvp[CAT_];
  __shared__ float red[256];
  const int bs  = blockIdx.x;            // b*S + s
  const int s   = bs & (S_ - 1);
  const int tid = threadIdx.x;

  for (int d = tid; d < DIM_; d += 256) xs[d] = x[(size_t)bs * DIM_ + d];
  __syncthreads();

  for (int j = tid; j < CAT_; j += 256) {
    const float* wr = wkv_a + (size_t)j * DIM_;
    float acc = 0.f;
    for (int d = 0; d < DIM_; ++d) acc += xs[d] * wr[d];
    kvp[j] = acc;
  }
  __syncthreads();

  float p = 0.f;
  for (int c = tid; c < KVR_; c += 256) p += kvp[c] * kvp[c];
  red[tid] = p;
  __syncthreads();
  for (int off = 128; off > 0; off >>= 1) {
    if (tid < off) red[tid] += red[tid + off];
    __syncthreads();
  }
  const float rms = rsqrtf(red[0] * (1.f / (float)KVR_) + 1e-6f);

  for (int c = tid; c < KVR_; c += 256)
    nkv[(size_t)bs * KVR_ + c] = kvp[c] * rms * kvnw[c];

  if (tid < DR_ / 2) {
    const int i = tid;
    const float re = kvp[KVR_ + 2 * i];
    const float im = kvp[KVR_ + 2 * i + 1];
    const float c  = fcos[s * (DR_ / 2) + i];
    const float sn = fsin[s * (DR_ / 2) + i];
    npe[(size_t)bs * DR_ + 2 * i]     = re * c - im * sn;
    npe[(size_t)bs * DR_ + 2 * i + 1] = re * sn + im * c;
  }
}

// ============ Kernel B: q projection + rope(q_pe) + absorb, store qT[bs][k][h] ============
__global__ void mla_q_proj(const float* __restrict__ x,
                           const float* __restrict__ wq,
                           const float* __restrict__ wkv_b,
                           const float* __restrict__ fcos,
                           const float* __restrict__ fsin,
                           float* __restrict__ qT) {
  __shared__ float xs[DIM_];
  __shared__ float qrow[H_ * QROW_];     // 3072
  const int bs  = blockIdx.x;
  const int s   = bs & (S_ - 1);
  const int tid = threadIdx.x;
  const float scale = rsqrtf((float)(DN_ + DR_));

  for (int d = tid; d < DIM_; d += 256) xs[d] = x[(size_t)bs * DIM_ + d];
  __syncthreads();

  for (int j = tid; j < H_ * QROW_; j += 256) {
    const float* wr = wq + (size_t)j * DIM_;
    float acc = 0.f;
    for (int d = 0; d < DIM_; ++d) acc += xs[d] * wr[d];
    qrow[j] = acc;
  }
  __syncthreads();

  // rope on q_pe part
  for (int idx = tid; idx < H_ * (DR_ / 2); idx += 256) {
    const int h = idx >> 5, i = idx & 31;
    const float re = qrow[h * QROW_ + DN_ + 2 * i];
    const float im = qrow[h * QROW_ + DN_ + 2 * i + 1];
    const float c  = fcos[s * 32 + i];
    const float sn = fsin[s * 32 + i];
    qrow[h * QROW_ + DN_ + 2 * i]     = re * c - im * sn;
    qrow[h * QROW_ + DN_ + 2 * i + 1] = re * sn + im * c;
  }
  __syncthreads();

  float* qTb = qT + (size_t)bs * CAT_ * H_;
  // absorbed part: q_abs[h][c] = sum_d q_nope[h][d] * wkv_b[h*256+d][c]
  for (int idx = tid; idx < H_ * KVR_; idx += 256) {
    const int h = idx >> 9, c = idx & (KVR_ - 1);
    const float* qh = qrow + h * QROW_;
    float acc = 0.f;
    for (int d = 0; d < DN_; ++d)
      acc += qh[d] * wkv_b[((size_t)(h * (DN_ + DV_) + d)) * KVR_ + c];
    qTb[c * H_ + h] = acc * scale;
  }
  // rope'd pe part
  for (int idx = tid; idx < H_ * DR_; idx += 256) {
    const int h = idx >> 6, r = idx & (DR_ - 1);
    qTb[(KVR_ + r) * H_ + h] = qrow[h * QROW_ + DN_ + r] * scale;
  }
}

// ============ Kernel C: flash attention over T (split-T), WMMA fp32, async LDS ============
__device__ __forceinline__ void issue_tile_load(
    float (*buf)[KPITCH_], int tbase, int b,
    const float* __restrict__ kvc, const float* __restrict__ pec,
    const float* __restrict__ nkv, const float* __restrict__ npe, int tid) {
  for (int idx = tid; idx < F4_PER_TILE_; idx += 128) {
    const int r   = idx / (CAT_ / 4);
    const int qw  = idx - r * (CAT_ / 4);
    const int col = qw * 4;
    const int t   = tbase + r;
    const float* src;
    if (col < KVR_) {
      src = (t < START_) ? kvc + ((size_t)b * T_ + t) * KVR_ + col
                         : nkv + ((size_t)(b * S_ + (t - START_))) * KVR_ + col;
    } else {
      const int pc = col - KVR_;
      src = (t < START_) ? pec + ((size_t)b * T_ + t) * DR_ + pc
                         : npe + ((size_t)(b * S_ + (t - START_))) * DR_ + pc;
    }
    copy16_lds(src, &buf[r][col]);
  }
}

__global__ void __launch_bounds__(128) mla_flash(
    const float* __restrict__ qT, const float* __restrict__ kv_cache,
    const float* __restrict__ pe_cache, const float* __restrict__ nkv,
    const float* __restrict__ npe, float* __restrict__ opart,
    float* __restrict__ mlpart) {
  __shared__ float buf0[TILE_][KPITCH_];  // double-buffered kv||pe tile
  __shared__ float buf1[TILE_][KPITCH_];
  __shared__ float qls[CAT_ * H_];        // staged qT[k][h] for this (b,s)
  __shared__ float scrp[4][H_][17];       // per-wave partial score tiles
  __shared__ float ps[TILE_][H_];         // probs transposed [t][h]
  __shared__ float fac[H_];               // per-head online-softmax rescale

  const int chunk = blockIdx.x, s = blockIdx.y, b = blockIdx.z;
  const int bs   = b * S_ + s;
  const int tid  = threadIdx.x;
  const int lane = tid & 31, wave = tid >> 5;
  const int hn   = lane & 15;             // col index (N) / head index for A
  const int koff = (lane >> 4) * 2;       // K offset for A/B halves
  const int hi   = (lane >> 4) * 8;       // M offset for C/D halves
  const float* qTb = qT + (size_t)bs * CAT_ * H_;

  v8f accO[8];                            // 16h x 128c per wave, 8 tiles
#pragma unroll
  for (int j = 0; j < 8; ++j)
#pragma unroll
    for (int e = 0; e < 8; ++e) accO[j][e] = 0.f;
  float m_run = -1e30f, l_run = 0.f;      // live in wave 0, lanes 0..15

  // stage q once (async), then first tile (async)
  for (int idx = tid; idx < (CAT_ * H_) / 4; idx += 128)
    copy16_lds(qTb + idx * 4, qls + idx * 4);
  issue_tile_load(buf0, chunk * CHUNK_T_, b, kv_cache, pe_cache, nkv, npe, tid);

  const int NT = CHUNK_T_ / TILE_;        // 32 tiles
  for (int tile = 0; tile < NT; ++tile) {
    float (*cur)[KPITCH_] = (tile & 1) ? buf1 : buf0;
    float (*nxt)[KPITCH_] = (tile & 1) ? buf0 : buf1;
    const int tbase = chunk * CHUNK_T_ + tile * TILE_;

    __syncthreads();                      // prev PV done before nxt is overwritten
    const bool pre = (tile + 1 < NT);
    if (pre)
      issue_tile_load(nxt, tbase + TILE_, b, kv_cache, pe_cache, nkv, npe, tid);
    wait_async(pre);                      // q + cur complete for this wave
    __syncthreads();                      // cur visible to all waves

    // phase 1: QK^T, all 4 waves, K split 4x144 (36 wmma each)
    {
      v8f acc;
#pragma unroll
      for (int e = 0; e < 8; ++e) acc[e] = 0.f;
      const int k0 = wave * (CAT_ / 4);   // 144 K per wave
      const float* kvl  = &cur[hn][koff];
      const float* qlsl = qls + koff * H_ + hn;
      for (int k = k0; k < k0 + CAT_ / 4; k += 4) {
        v2f a, bv;
        a.x  = qlsl[k * H_];              // A[h][k+koff]
        a.y  = qlsl[k * H_ + H_];         // A[h][k+koff+1]
        bv.x = kvl[k];                    // B[k+koff][t=hn]
        bv.y = kvl[k + 1];
        acc = wmma4(a, bv, acc);
      }
#pragma unroll
      for (int e = 0; e < 8; ++e) scrp[wave][e + hi][hn] = acc[e];
    }
    __syncthreads();

    // online softmax stats: wave0, lane = head row; reduce the 4 K-partials
    if (wave == 0 && lane < 16) {
      const int h = lane;
      float rowv[16];
      float rm = -1e30f;
#pragma unroll
      for (int n = 0; n < 16; ++n) {
        float sv = scrp[0][h][n] + scrp[1][h][n] + scrp[2][h][n] + scrp[3][h][n];
        const int tg = tbase + n;
        if (tg > START_ + s) sv += NEG_;  // causal mask
        rowv[n] = sv;
        rm = fmaxf(rm, sv);
      }
      const float nm = fmaxf(m_run, rm);
      const float f  = __expf(m_run - nm);
      float psum = 0.f;
#pragma unroll
      for (int n = 0; n < 16; ++n) {
        const float pv = __expf(rowv[n] - nm);
        ps[n][h] = pv;
        psum += pv;
      }
      l_run = l_run * f + psum;
      m_run = nm;
      fac[h] = f;
    }
    __syncthreads();

    // phase 2: PV, each wave owns c-block [wave*128, wave*128+128)
    {
      float fr[8];
#pragma unroll
      for (int e = 0; e < 8; ++e) fr[e] = fac[e + hi];
#pragma unroll
      for (int j = 0; j < 8; ++j)
#pragma unroll
        for (int e = 0; e < 8; ++e) accO[j][e] *= fr[e];

      const int c0 = wave * 128;
      for (int k = 0; k < TILE_; k += 4) {
        v2f a;
        a.x = ps[k + koff][hn];           // A[h][t=k+koff]
        a.y = ps[k + koff + 1][hn];
#pragma unroll
        for (int j = 0; j < 8; ++j) {
          v2f bv;
          bv.x = cur[k + koff][c0 + j * 16 + hn];       // B[t][c]
          bv.y = cur[k + koff + 1][c0 + j * 16 + hn];
          accO[j] = wmma4(a, bv, accO[j]);
        }
      }
    }
  }

  // write split-T partials
  {
    const size_t obase = ((size_t)(bs * NCHUNK_ + chunk)) * H_ * KVR_;
    const int c0 = wave * 128;
#pragma unroll
    for (int j = 0; j < 8; ++j)
#pragma unroll
      for (int e = 0; e < 8; ++e) {
        const int h = e + hi;
        const int c = c0 + j * 16 + hn;
        opart[obase + (size_t)h * KVR_ + c] = accO[j][e];
      }
    if (wave == 0 && lane < 16) {
      const size_t mb = ((size_t)(bs * NCHUNK_ + chunk)) * H_ * 2 + lane * 2;
      mlpart[mb]     = m_run;
      mlpart[mb + 1] = l_run;
    }
  }
}

// ============ Kernel D: combine split-T partials + V-absorb ============
__global__ void mla_combine(const float* __restrict__ opart,
                            const float* __restrict__ mlpart,
                            const float* __restrict__ wkv_b,
                            float* __restrict__ oproj) {
  __shared__ float ov[KVR_];
  const int bsh = blockIdx.x;
  const int bs = bsh >> 4, h = bsh & 15;
  const int tid = threadIdx.x;            // 128 threads

  float mi[NCHUNK_], li[NCHUNK_], wgt[NCHUNK_];
  float M = -1e30f;
#pragma unroll
  for (int i = 0; i < NCHUNK_; ++i) {
    const size_t mb = ((size_t)(bs * NCHUNK_ + i)) * H_ * 2 + h * 2;
    mi[i] = mlpart[mb];
    li[i] = mlpart[mb + 1];
    M = fmaxf(M, mi[i]);
  }
  float L = 0.f;
#pragma unroll
  for (int i = 0; i < NCHUNK_; ++i) {
    wgt[i] = __expf(mi[i] - M);
    L += li[i] * wgt[i];
  }
  const float invL = 1.f / L;

  for (int c = tid; c < KVR_; c += 128) {
    float acc = 0.f;
#pragma unroll
    for (int i = 0; i < NCHUNK_; ++i)
      acc += wgt[i] * opart[((size_t)(bs * NCHUNK_ + i) * H_ + h) * KVR_ + c];
    ov[c] = acc * invL;
  }
  __syncthreads();

  // V-absorb: o2[d] = sum_c o[c] * wkv_b[h*256 + 128 + d][c]
  for (int d = tid; d < DV_; d += 128) {
    const float* wr = wkv_b + ((size_t)(h * (DN_ + DV_) + DN_ + d)) * KVR_;
    float acc = 0.f;
    for (int c = 0; c < KVR_; ++c) acc += ov[c] * wr[c];
    oproj[(size_t)bs * (H_ * DV_) + h * DV_ + d] = acc;
  }
}

// ============ Kernel E: output projection ============
__global__ void mla_out_proj(const float* __restrict__ oproj,
                             const float* __restrict__ wo,
                             float* __restrict__ out) {
  __shared__ float ol[DIM_];
  const int bs = blockIdx.x;
  const int tid = threadIdx.x;            // 256 threads
  for (int k = tid; k < DIM_; k += 256) ol[k] = oproj[(size_t)bs * DIM_ + k];
  __syncthreads();
  for (int j = tid; j < DIM_; j += 256) {
    const float* wr = wo + (size_t)j * DIM_;
    float acc = 0.f;
    for (int k = 0; k < DIM_; ++k) acc += ol[k] * wr[k];
    out[(size_t)bs * DIM_ + j] = acc;
  }
}

// ============ host launcher ============
extern "C" void kernel_launch(void* const* d_in, const int* in_sizes, int n_in,
                              void* d_out, int out_size, void* d_ws, size_t ws_size,
                              hipStream_t stream) {
  const float* x    = (const float*)d_in[0];
  const float* wq   = (const float*)d_in[1];
  const float* wkva = (const float*)d_in[2];
  const float* kvnw = (const float*)d_in[3];
  const float* wkvb = (const float*)d_in[4];
  const float* wo   = (const float*)d_in[5];
  const float* kvc  = (const float*)d_in[6];
  const float* pec  = (const float*)d_in[7];
  const float* fc   = (const float*)d_in[8];
  const float* fs   = (const float*)d_in[9];
  // d_in[10] (mask) and d_in[11] (start_pos) folded into constants

  float* ws    = (float*)d_ws;
  float* qT    = ws + QT_OFF;
  float* nkv   = ws + NKV_OFF;
  float* npe   = ws + NPE_OFF;
  float* opart = ws + OPART_OFF;
  float* mlp   = ws + ML_OFF;
  float* oproj = ws + OPROJ_OFF;

  float* out = (float*)d_out;

  mla_kv_prep<<<B_ * S_, 256, 0, stream>>>(x, wkva, kvnw, fc, fs, nkv, npe);
  mla_q_proj<<<B_ * S_, 256, 0, stream>>>(x, wq, wkvb, fc, fs, qT);
  mla_flash<<<dim3(NCHUNK_, S_, B_), 128, 0, stream>>>(qT, kvc, pec, nkv, npe,
                                                       opart, mlp);
  mla_combine<<<B_ * S_ * H_, 128, 0, stream>>>(opart, mlp, wkvb, oproj);
  mla_out_proj<<<B_ * S_, 256, 0, stream>>>(oproj, wo, out);
}